// ProbAttention_31550829756768
// MI455X (gfx1250) — compile-verified
//
#include <hip/hip_runtime.h>
#include <math.h>

typedef __attribute__((ext_vector_type(2))) float v2f;
typedef __attribute__((ext_vector_type(8))) float v8f;

#define NB   4
#define NN   1024
#define DD   256
#define NCLS 10
#define UU   140
#define UPAD 144

// ---------------------------------------------------------------------------
// K0: fused projections  O = X @ W^T (+bias on the Wadd branch)
// X:(B*N,256) row-major, W:(256,256) row-major. One wave per 16x16 tile.
// WMMA f32 16x16x4: A lane l -> A[l&15][kb+2*hi+{0,1}], B lane l -> B[kb+2*hi+{0,1}][l&15]
// ---------------------------------------------------------------------------
__global__ __launch_bounds__(32)
void k_proj(const float* __restrict__ X,
            const float* __restrict__ Wq, const float* __restrict__ Wk,
            const float* __restrict__ Wv, const float* __restrict__ Wa,
            const float* __restrict__ badd,
            float* __restrict__ Q, float* __restrict__ K,
            float* __restrict__ V, float* __restrict__ A)
{
    const int lane = threadIdx.x;
    const int r16  = lane & 15;
    const int hi   = lane >> 4;
    const int mt   = blockIdx.x;      // 0..255
    const int nt   = blockIdx.y;      // 0..15
    const int which = blockIdx.z;     // 0..3

    const float* W; float* O;
    switch (which) {
      case 0:  W = Wq; O = Q; break;
      case 1:  W = Wk; O = K; break;
      case 2:  W = Wv; O = V; break;
      default: W = Wa; O = A; break;
    }

    const int m0 = mt * 16;
    const int n0 = nt * 16;
    const float* xrow = X + (size_t)(m0 + r16) * DD;   // A row for this lane
    const float* wrow = W + (size_t)(n0 + r16) * DD;   // B col n = W row n (B = W^T)

    v8f acc = {};
    for (int kb = 0; kb < DD; kb += 4) {
        v2f a = *(const v2f*)(xrow + kb + 2 * hi);
        v2f b = *(const v2f*)(wrow + kb + 2 * hi);
        acc = __builtin_amdgcn_wmma_f32_16x16x4_f32(false, a, false, b,
                                                    (short)0, acc, false, false);
    }
    const float bias = (which == 3) ? badd[n0 + r16] : 0.0f;
#pragma unroll
    for (int r = 0; r < 8; ++r) {
        const int m = m0 + r + 8 * hi;
        O[(size_t)m * DD + n0 + r16] = acc[r] + bias;
    }
}

// ---------------------------------------------------------------------------
// K1: sparse sampling score  M[b,q] = max_s(Q.K_s) - sum_s(Q.K_s)/N
// one wave per (b,q); lane covers k = lane + 32*j; butterfly reduce
// ---------------------------------------------------------------------------
__global__ __launch_bounds__(256)
void k_sample(const float* __restrict__ Q, const float* __restrict__ K,
              const int* __restrict__ idx, float* __restrict__ Msc)
{
    const int wave = (int)((blockIdx.x * blockDim.x + threadIdx.x) >> 5);
    const int lane = threadIdx.x & 31;
    if (wave >= NB * NN) return;
    const int b = wave / NN;
    const int q = wave % NN;

    const float* qp = Q + (size_t)(b * NN + q) * DD;
    float qr[8];
#pragma unroll
    for (int j = 0; j < 8; ++j) qr[j] = qp[lane + 32 * j];

    float mx = -INFINITY, sm = 0.0f;
    for (int s = 0; s < UU; ++s) {
        const int ks = idx[q * UU + s];
        const float* kp = K + (size_t)(b * NN + ks) * DD;
        float p = 0.0f;
#pragma unroll
        for (int j = 0; j < 8; ++j) p += qr[j] * kp[lane + 32 * j];
#pragma unroll
        for (int off = 16; off > 0; off >>= 1)
            p += __shfl_xor(p, off, 32);
        mx = fmaxf(mx, p);
        sm += p;
    }
    if (lane == 0) Msc[wave] = mx - sm * (1.0f / (float)NN);
}

// ---------------------------------------------------------------------------
// K2: per-batch top-140 (iterative argmax, lower index wins ties like lax.top_k)
// ---------------------------------------------------------------------------
__global__ __launch_bounds__(256)
void k_topk(const float* __restrict__ Msc, int* __restrict__ Mtop)
{
    __shared__ float buf[NN];
    __shared__ float rv[256];
    __shared__ int   ri[256];
    const int b = blockIdx.x;
    const int t = threadIdx.x;
    for (int i = t; i < NN; i += 256) buf[i] = Msc[b * NN + i];
    __syncthreads();
    for (int it = 0; it < UU; ++it) {
        float bv = -INFINITY; int bi = NN;
        for (int i = t; i < NN; i += 256) {
            const float v = buf[i];
            if (v > bv || (v == bv && i < bi)) { bv = v; bi = i; }
        }
        rv[t] = bv; ri[t] = bi;
        __syncthreads();
        for (int off = 128; off > 0; off >>= 1) {
            if (t < off) {
                const float v2 = rv[t + off]; const int i2 = ri[t + off];
                if (v2 > rv[t] || (v2 == rv[t] && i2 < ri[t])) { rv[t] = v2; ri[t] = i2; }
            }
            __syncthreads();
        }
        if (t == 0) { Mtop[b * UU + it] = ri[0]; buf[ri[0]] = -INFINITY; }
        __syncthreads();
    }
}

// ---------------------------------------------------------------------------
// K3: scores = Q_red @ K^T * (1/16)   -> S[b, m(0..143), k(0..1023)]
// ---------------------------------------------------------------------------
__global__ __launch_bounds__(32)
void k_scores(const float* __restrict__ Q, const float* __restrict__ K,
              const int* __restrict__ Mtop, float* __restrict__ S)
{
    const int lane = threadIdx.x, r16 = lane & 15, hi = lane >> 4;
    const int mt = blockIdx.x;   // 0..8
    const int nt = blockIdx.y;   // 0..63
    const int b  = blockIdx.z;

    const int mrow = mt * 16 + r16;
    const int qrow = (mrow < UU) ? Mtop[b * UU + mrow] : 0;
    const float* qp = Q + (size_t)(b * NN + qrow) * DD;
    const float* kp = K + (size_t)(b * NN + nt * 16 + r16) * DD;

    v8f acc = {};
    for (int kb = 0; kb < DD; kb += 4) {
        v2f a  = *(const v2f*)(qp + kb + 2 * hi);
        v2f bb = *(const v2f*)(kp + kb + 2 * hi);
        acc = __builtin_amdgcn_wmma_f32_16x16x4_f32(false, a, false, bb,
                                                    (short)0, acc, false, false);
    }
    const float scale = 0.0625f;  // 1/sqrt(256)
#pragma unroll
    for (int r = 0; r < 8; ++r) {
        const int m = mt * 16 + r + 8 * hi;
        S[((size_t)b * UPAD + m) * NN + nt * 16 + r16] = acc[r] * scale;
    }
}

// ---------------------------------------------------------------------------
// K4: row softmax over 1024 keys (only the 140 real rows)
// ---------------------------------------------------------------------------
__global__ __launch_bounds__(256)
void k_softmax(float* __restrict__ S)
{
    const int row = blockIdx.x;            // B*UU rows
    const int b = row / UU, m = row % UU;
    float* p = S + ((size_t)b * UPAD + m) * NN;
    const int t = threadIdx.x;
    __shared__ float red[256];

    float mx = -INFINITY;
    for (int i = t; i < NN; i += 256) mx = fmaxf(mx, p[i]);
    red[t] = mx; __syncthreads();
    for (int off = 128; off > 0; off >>= 1) {
        if (t < off) red[t] = fmaxf(red[t], red[t + off]);
        __syncthreads();
    }
    mx = red[0]; __syncthreads();

    float sm = 0.0f;
    for (int i = t; i < NN; i += 256) { const float e = expf(p[i] - mx); p[i] = e; sm += e; }
    red[t] = sm; __syncthreads();
    for (int off = 128; off > 0; off >>= 1) {
        if (t < off) red[t] += red[t + off];
        __syncthreads();
    }
    const float inv = 1.0f / red[0];
    for (int i = t; i < NN; i += 256) p[i] *= inv;
}

// ---------------------------------------------------------------------------
// K5: out_top = attn @ V   (144x256 per batch, K=1024)
// ---------------------------------------------------------------------------
__global__ __launch_bounds__(32)
void k_av(const float* __restrict__ S, const float* __restrict__ V,
          float* __restrict__ OT)
{
    const int lane = threadIdx.x, r16 = lane & 15, hi = lane >> 4;
    const int mt = blockIdx.x;  // 0..8
    const int nt = blockIdx.y;  // 0..15
    const int b  = blockIdx.z;

    const float* ap = S + ((size_t)b * UPAD + mt * 16 + r16) * NN;
    const float* vb = V + (size_t)b * NN * DD + nt * 16 + r16;   // column n for this lane

    v8f acc = {};
    for (int kb = 0; kb < NN; kb += 4) {
        v2f a = *(const v2f*)(ap + kb + 2 * hi);
        v2f bb;
        bb.x = vb[(size_t)(kb + 2 * hi    ) * DD];
        bb.y = vb[(size_t)(kb + 2 * hi + 1) * DD];
        acc = __builtin_amdgcn_wmma_f32_16x16x4_f32(false, a, false, bb,
                                                    (short)0, acc, false, false);
    }
#pragma unroll
    for (int r = 0; r < 8; ++r) {
        const int m = mt * 16 + r + 8 * hi;
        OT[((size_t)b * UPAD + m) * DD + nt * 16 + r16] = acc[r];
    }
}

// ---------------------------------------------------------------------------
// K6: V column means per batch
// ---------------------------------------------------------------------------
__global__ __launch_bounds__(256)
void k_vmean(const float* __restrict__ V, float* __restrict__ VM)
{
    const int b = blockIdx.x, d = threadIdx.x;
    float s = 0.0f;
    for (int n = 0; n < NN; ++n) s += V[((size_t)b * NN + n) * DD + d];
    VM[b * DD + d] = s * (1.0f / (float)NN);
}

// ---------------------------------------------------------------------------
// K7: context = Vmean (broadcast) + residual projection
// ---------------------------------------------------------------------------
__global__ __launch_bounds__(256)
void k_ctx(const float* __restrict__ VM, const float* __restrict__ A,
           float* __restrict__ C)
{
    const size_t i = (size_t)blockIdx.x * 256 + threadIdx.x;  // B*N*D elements
    const int d = (int)(i & 255);
    const int b = (int)(i >> 18);                             // / (1024*256)
    C[i] = VM[b * DD + d] + A[i];
}

// ---------------------------------------------------------------------------
// K8: scatter attention rows into context (top-k indices are unique)
// ---------------------------------------------------------------------------
__global__ __launch_bounds__(256)
void k_scatter(const int* __restrict__ Mtop, const float* __restrict__ OT,
               const float* __restrict__ A, float* __restrict__ C)
{
    const int b = blockIdx.x / UU, m = blockIdx.x % UU, d = threadIdx.x;
    const int n = Mtop[b * UU + m];
    C[((size_t)b * NN + n) * DD + d] =
        OT[((size_t)b * UPAD + m) * DD + d] + A[((size_t)b * NN + n) * DD + d];
}

// ---------------------------------------------------------------------------
// K9: classifier head  out[b,c] = context[b,:] . Wfin[c,:] + bfin[c]
// ---------------------------------------------------------------------------
__global__ __launch_bounds__(256)
void k_final(const float* __restrict__ C, const float* __restrict__ Wf,
             const float* __restrict__ bf, float* __restrict__ out)
{
    const int b = blockIdx.x / NCLS, c = blockIdx.x % NCLS, t = threadIdx.x;
    const size_t L = (size_t)NN * DD;  // 262144
    const float* cp = C  + (size_t)b * L;
    const float* wp = Wf + (size_t)c * L;
    float s = 0.0f;
    for (size_t j = t; j < L; j += 256) s += cp[j] * wp[j];
    __shared__ float red[256];
    red[t] = s; __syncthreads();
    for (int off = 128; off > 0; off >>= 1) {
        if (t < off) red[t] += red[t + off];
        __syncthreads();
    }
    if (t == 0) out[b * NCLS + c] = red[0] + bf[c];
}

// ---------------------------------------------------------------------------
extern "C" void kernel_launch(void* const* d_in, const int* in_sizes, int n_in,
                              void* d_out, int out_size, void* d_ws, size_t ws_size,
                              hipStream_t stream)
{
    (void)in_sizes; (void)n_in; (void)out_size; (void)ws_size;

    const float* inp  = (const float*)d_in[0];
    // d_in[1..6] (fai_x, fai_x_prime, w_1, b_1, w_2, b_2) are unused by the reference
    const float* Wq   = (const float*)d_in[7];
    const float* Wk   = (const float*)d_in[8];
    const float* Wv   = (const float*)d_in[9];
    const float* Wa   = (const float*)d_in[10];
    const float* badd = (const float*)d_in[11];
    const float* Wfin = (const float*)d_in[12];
    const float* bfin = (const float*)d_in[13];
    const int*   idx  = (const int*)d_in[14];
    float* out = (float*)d_out;

    // workspace layout (floats)
    float* Q    = (float*)d_ws;                 // 1048576
    float* K    = Q + (size_t)NB * NN * DD;     // 1048576
    float* V    = K + (size_t)NB * NN * DD;     // 1048576
    float* A    = V + (size_t)NB * NN * DD;     // 1048576
    float* Msc  = A + (size_t)NB * NN * DD;     // 4096
    int*   Mtop = (int*)(Msc + NB * NN);        // 560 -> pad 1024
    float* S    = (float*)(Mtop + 1024);        // 4*144*1024 = 589824
    float* OT   = S + (size_t)NB * UPAD * NN;   // 4*144*256  = 147456
    float* VM   = OT + (size_t)NB * UPAD * DD;  // 1024
    float* C    = VM + NB * DD;                 // 1048576

    k_proj   <<<dim3((NB * NN) / 16, DD / 16, 4), 32, 0, stream>>>(inp, Wq, Wk, Wv, Wa, badd, Q, K, V, A);
    k_sample <<<dim3((NB * NN * 32) / 256), 256, 0, stream>>>(Q, K, idx, Msc);
    k_topk   <<<NB, 256, 0, stream>>>(Msc, Mtop);
    k_scores <<<dim3(UPAD / 16, NN / 16, NB), 32, 0, stream>>>(Q, K, Mtop, S);
    k_softmax<<<NB * UU, 256, 0, stream>>>(S);
    k_av     <<<dim3(UPAD / 16, DD / 16, NB), 32, 0, stream>>>(S, V, OT);
    k_vmean  <<<NB, 256, 0, stream>>>(V, VM);
    k_ctx    <<<(NB * NN * DD) / 256, 256, 0, stream>>>(VM, A, C);
    k_scatter<<<NB * UU, 256, 0, stream>>>(Mtop, OT, A, C);
    k_final  <<<NB * NCLS, 256, 0, stream>>>(C, Wfin, bfin, out);
}